// GraphModel_21062519619885
// MI455X (gfx1250) — compile-verified
//
#include <hip/hip_runtime.h>

typedef __attribute__((ext_vector_type(2))) float v2f;
typedef __attribute__((ext_vector_type(8))) float v8f;

#define DH 128
#define DOUT 64

__device__ __forceinline__ void atomic_add_f32(float* p, float v) {
    // hardware global_atomic_add_f32 (no CAS loop)
    unsafeAtomicAdd(p, v);
}

// ---------------- degree / normalization ----------------

__global__ void init_deg_kernel(float* dis, int N) {
    int i = blockIdx.x * blockDim.x + threadIdx.x;
    if (i < N) dis[i] = 1.0f;  // self-loop contribution
}

__global__ void deg_count_kernel(const int* __restrict__ dst, float* dis, int E) {
    int e = blockIdx.x * blockDim.x + threadIdx.x;
    if (e < E) atomic_add_f32(&dis[dst[e]], 1.0f);
}

__global__ void deg_finalize_kernel(float* dis, int N) {
    int i = blockIdx.x * blockDim.x + threadIdx.x;
    if (i < N) dis[i] = rsqrtf(dis[i]);  // deg >= 1 always (self-loops)
}

// ---------------- WMMA fp32 GEMM with LDS-staged operands ----------------
// C[M x NCOLS] = A[M x 128] * B[128 x NCOLS] (+bias)
// Block covers 32 rows x NCOLS cols: 2 row-tiles x (NCOLS/32) col-waves,
// each wave owns 2 16x16 output tiles. blockDim = 64 * (NCOLS/32) * ... = 2*NCOLS.
// B is staged transposed Bs[col][k] (stride K+4) so a WMMA B fragment is one
// aligned ds_load_b64; A strip staged the same way. Fully unrolled K loop.

template <int NCOLS>
__global__ void gemm_wmma_f32_lds(const float* __restrict__ A, const float* __restrict__ B,
                                  const float* __restrict__ bias, float* __restrict__ C,
                                  int M) {
    constexpr int K = 128;
    constexpr int LDST = K + 4;        // 132: 16 lanes x 2 dwords -> 64 distinct banks
    constexpr int ROWS = 32;           // rows per block
    constexpr int CW = NCOLS / 32;     // col-waves per row tile

    extern __shared__ float smem[];
    float* Bs = smem;                   // [NCOLS][LDST]
    float* As = smem + NCOLS * LDST;    // [ROWS][LDST]

    const int tid = threadIdx.x;
    const int nthr = blockDim.x;
    const int row0 = blockIdx.x * ROWS;

    // stage B: global row-major [k][col] -> LDS [col][k]
    for (int idx = tid; idx < K * (NCOLS / 4); idx += nthr) {
        int k  = idx / (NCOLS / 4);
        int c4 = (idx % (NCOLS / 4)) * 4;
        float4 v = ((const float4*)B)[idx];
        Bs[(c4 + 0) * LDST + k] = v.x;
        Bs[(c4 + 1) * LDST + k] = v.y;
        Bs[(c4 + 2) * LDST + k] = v.z;
        Bs[(c4 + 3) * LDST + k] = v.w;
    }
    // stage A strip (rows clamped; OOB rows masked at store)
    for (int idx = tid; idx < ROWS * (K / 4); idx += nthr) {
        int r  = idx / (K / 4);
        int k4 = (idx % (K / 4)) * 4;
        int gr = row0 + r;
        if (gr >= M) gr = M - 1;
        float4 v = *(const float4*)(A + (size_t)gr * K + k4);
        float* ap = As + r * LDST + k4;
        ap[0] = v.x; ap[1] = v.y; ap[2] = v.z; ap[3] = v.w;
    }
    __syncthreads();

    const int wave = tid >> 5;
    const int lane = tid & 31;
    const int rt   = wave / CW;            // local row tile 0..1
    const int ct0  = (wave % CW) * 2;      // first of 2 col tiles
    const int m    = lane & 15;
    const int koff = (lane >> 4) * 2;      // lanes 0-15: K {0,1}; 16-31: K {2,3}

    const float* arow = As + (rt * 16 + m) * LDST + koff;
    const float* b0   = Bs + ((ct0 + 0) * 16 + m) * LDST + koff;
    const float* b1   = Bs + ((ct0 + 1) * 16 + m) * LDST + koff;

    v8f acc0 = (v8f){0.f,0.f,0.f,0.f,0.f,0.f,0.f,0.f};
    v8f acc1 = (v8f){0.f,0.f,0.f,0.f,0.f,0.f,0.f,0.f};

#pragma unroll
    for (int k0 = 0; k0 < K; k0 += 4) {
        v2f a   = *(const v2f*)(arow + k0);
        v2f bb0 = *(const v2f*)(b0 + k0);
        v2f bb1 = *(const v2f*)(b1 + k0);
        acc0 = __builtin_amdgcn_wmma_f32_16x16x4_f32(false, a, false, bb0, (short)0, acc0, false, false);
        acc1 = __builtin_amdgcn_wmma_f32_16x16x4_f32(false, a, false, bb1, (short)0, acc1, false, false);
    }

    // D layout: VGPR r -> M=r (lanes 0-15), M=r+8 (lanes 16-31); N = lane&15
    const int rbase = row0 + rt * 16 + (lane >> 4) * 8;
    const int ncol  = lane & 15;
    const int col0  = (ct0 + 0) * 16 + ncol;
    const int col1  = (ct0 + 1) * 16 + ncol;
    const float bv0 = bias ? bias[col0] : 0.0f;
    const float bv1 = bias ? bias[col1] : 0.0f;
#pragma unroll
    for (int r = 0; r < 8; ++r) {
        int rr = rbase + r;
        if (rr < M) {
            C[(size_t)rr * NCOLS + col0] = acc0[r] + bv0;
            C[(size_t)rr * NCOLS + col1] = acc1[r] + bv1;
        }
    }
}

// ---------------- aggregation ----------------

// agg[n][:] = xw[n][:] * dis[n]^2   (self-loop term; also initializes agg)
__global__ void init_agg_kernel(const float* __restrict__ xw, const float* __restrict__ dis,
                                float* __restrict__ agg, int N) {
    size_t i = (size_t)blockIdx.x * blockDim.x + threadIdx.x;  // over N*32 float4 groups
    if (i >= (size_t)N * (DH / 4)) return;
    int node = (int)(i >> 5);
    float d = dis[node];
    float w = d * d;
    float4 v = ((const float4*)xw)[i];
    float4 o;
    o.x = v.x * w; o.y = v.y * w; o.z = v.z * w; o.w = v.w * w;
    ((float4*)agg)[i] = o;
}

// One wave per edge: lane handles 4 contiguous dims. float4 gather + 4 f32 atomics.
__global__ void scatter_edges_kernel(const float* __restrict__ xw, const int* __restrict__ src,
                                     const int* __restrict__ dst, const float* __restrict__ dis,
                                     float* __restrict__ agg, int E) {
    const int wavesPerBlock = blockDim.x >> 5;
    const int e = blockIdx.x * wavesPerBlock + (threadIdx.x >> 5);
    if (e >= E) return;
    const int lane = threadIdx.x & 31;
    const int s = src[e];
    const int d = dst[e];
    const float w = dis[s] * dis[d];
    float4 v = *(const float4*)(xw + (size_t)s * DH + lane * 4);
    float* arow = agg + (size_t)d * DH + lane * 4;
    atomic_add_f32(arow + 0, v.x * w);
    atomic_add_f32(arow + 1, v.y * w);
    atomic_add_f32(arow + 2, v.z * w);
    atomic_add_f32(arow + 3, v.w * w);
}

// ---------------- fused bias + LayerNorm + ReLU + residual (h += y) ----------------
// One wave per node, lane holds 4 dims; wave32 shuffle reductions.
__global__ void ln_relu_res_kernel(const float* __restrict__ agg, const float* __restrict__ gb,
                                   const float* __restrict__ g, const float* __restrict__ b,
                                   float* __restrict__ h, int N) {
    const int wavesPerBlock = blockDim.x >> 5;
    const int node = blockIdx.x * wavesPerBlock + (threadIdx.x >> 5);
    if (node >= N) return;
    const int lane = threadIdx.x & 31;

    float4 v  = *(const float4*)(agg + (size_t)node * DH + lane * 4);
    float4 bb = *(const float4*)(gb + lane * 4);
    v.x += bb.x; v.y += bb.y; v.z += bb.z; v.w += bb.w;

    float s = v.x + v.y + v.z + v.w;
#pragma unroll
    for (int off = 16; off; off >>= 1) s += __shfl_xor(s, off, 32);
    const float mu = s * (1.0f / DH);

    float dx = v.x - mu, dy = v.y - mu, dz = v.z - mu, dw = v.w - mu;
    float vs = dx * dx + dy * dy + dz * dz + dw * dw;
#pragma unroll
    for (int off = 16; off; off >>= 1) vs += __shfl_xor(vs, off, 32);
    const float inv = rsqrtf(vs * (1.0f / DH) + 1e-5f);

    float4 gg = *(const float4*)(g + lane * 4);
    float4 be = *(const float4*)(b + lane * 4);
    float* hp = h + (size_t)node * DH + lane * 4;
    float4 hv = *(float4*)hp;
    hv.x += fmaxf(dx * inv * gg.x + be.x, 0.0f);
    hv.y += fmaxf(dy * inv * gg.y + be.y, 0.0f);
    hv.z += fmaxf(dz * inv * gg.z + be.z, 0.0f);
    hv.w += fmaxf(dw * inv * gg.w + be.w, 0.0f);
    *(float4*)hp = hv;
}

// ---------------- host launcher ----------------

extern "C" void kernel_launch(void* const* d_in, const int* in_sizes, int n_in,
                              void* d_out, int out_size, void* d_ws, size_t ws_size,
                              hipStream_t stream) {
    const float* x     = (const float*)d_in[0];
    const int*   ei    = (const int*)d_in[1];
    const float* W_in  = (const float*)d_in[2];
    const float* b_in  = (const float*)d_in[3];
    const float* gcn_W = (const float*)d_in[4];
    const float* gcn_b = (const float*)d_in[5];
    const float* ln_g  = (const float*)d_in[6];
    const float* ln_b  = (const float*)d_in[7];
    const float* W_out = (const float*)d_in[8];
    const float* b_out = (const float*)d_in[9];

    const int N = in_sizes[0] / DH;   // DIN == DH == 128
    const int E = in_sizes[1] / 2;
    const int* src = ei;
    const int* dst = ei + E;

    char* ws = (char*)d_ws;
    const size_t fbytes = (size_t)N * DH * sizeof(float);
    float* h   = (float*)(ws);
    float* xw  = (float*)(ws + fbytes);
    float* agg = (float*)(ws + 2 * fbytes);
    float* dis = (float*)(ws + 3 * fbytes);

    // 1) symmetric normalization coefficients
    init_deg_kernel<<<(N + 255) / 256, 256, 0, stream>>>(dis, N);
    deg_count_kernel<<<(E + 255) / 256, 256, 0, stream>>>(dst, dis, E);
    deg_finalize_kernel<<<(N + 255) / 256, 256, 0, stream>>>(dis, N);

    const int blkTiles = (N + 31) / 32;
    const int smem128 = (128 + 32) * 132 * sizeof(float);  // 84.5 KB
    const int smem64  = (64 + 32) * 132 * sizeof(float);   // 50.7 KB

    // 2) h = x @ W_in + b_in
    gemm_wmma_f32_lds<128><<<blkTiles, 256, smem128, stream>>>(x, W_in, b_in, h, N);

    // 3) GCN layers
    for (int l = 0; l < 3; ++l) {
        gemm_wmma_f32_lds<128><<<blkTiles, 256, smem128, stream>>>(
            h, gcn_W + (size_t)l * DH * DH, nullptr, xw, N);
        init_agg_kernel<<<(int)(((size_t)N * (DH / 4) + 255) / 256), 256, 0, stream>>>(
            xw, dis, agg, N);
        scatter_edges_kernel<<<(E + 7) / 8, 256, 0, stream>>>(xw, src, dst, dis, agg, E);
        ln_relu_res_kernel<<<(N + 7) / 8, 256, 0, stream>>>(
            agg, gcn_b + (size_t)l * DH, ln_g + (size_t)l * DH, ln_b + (size_t)l * DH, h, N);
    }

    // 4) out = h @ W_out + b_out
    gemm_wmma_f32_lds<64><<<blkTiles, 128, smem64, stream>>>(h, W_out, b_out, (float*)d_out, N);
}